// FourierSurface_46119358824602
// MI455X (gfx1250) — compile-verified
//
#include <hip/hip_runtime.h>
#include <hip/hip_bf16.h>

#define ORDER 24
#define JK (ORDER * ORDER)          // 576
#define PTS_PER_WAVE 16
#define WAVES_PER_BLOCK 4
#define THREADS_PER_BLOCK 128
#define PTS_PER_BLOCK (PTS_PER_WAVE * WAVES_PER_BLOCK)   // 64

typedef __attribute__((ext_vector_type(2))) float v2f;
typedef __attribute__((ext_vector_type(8))) float v8f;

// Hardware V_SIN_F32: computes sin(2*pi*t) with internal range reduction,
// valid for |t| < 256 (here |t| <= 23). Exactly the reference's sin(2*pi*x*n).
__device__ __forceinline__ float sin2pi(float t) {
    return __builtin_amdgcn_sinf(t);
}

__launch_bounds__(THREADS_PER_BLOCK)
__global__ void FourierSurface_46119358824602_kernel(
        const float* __restrict__ xyz,
        const float* __restrict__ coeffs,
        float* __restrict__ out,
        int npts) {
    // Transposed coeffs: sCt[jk*24 + i] = C[i, jk].  +8 floats zero pad so
    // the (masked-to-zero) high-column loads stay in bounds.
    __shared__ float sCt[JK * ORDER + 8];
    // Sx per local point, +8 zero pad for the unconditional hi-column read.
    __shared__ float sSx[PTS_PER_BLOCK * ORDER + 8];

    const int t    = threadIdx.x;
    const int wave = t >> 5;
    const int lane = t & 31;
    const int half = lane >> 4;     // 0: lanes 0-15, 1: lanes 16-31
    const int lm   = lane & 15;

    // ---- stage transposed coefficients into LDS (coalesced global reads) ----
    for (int idx = t; idx < ORDER * JK; idx += THREADS_PER_BLOCK) {
        int i  = idx / JK;                  // coeffs[i*576 + jk]
        int jk = idx - i * JK;
        sCt[jk * ORDER + i] = coeffs[idx];
    }
    if (t < 8) {
        sCt[JK * ORDER + t]          = 0.0f;
        sSx[PTS_PER_BLOCK * ORDER + t] = 0.0f;
    }

    // ---- per-lane point and sine tables ----
    const int p_local  = wave * PTS_PER_WAVE + lm;
    const int p_global = blockIdx.x * PTS_PER_BLOCK + p_local;
    const int p_clamp  = (p_global < npts) ? p_global : (npts - 1);
    const float x = xyz[p_clamp * 3 + 0];
    const float y = xyz[p_clamp * 3 + 1];
    const float z = xyz[p_clamp * 3 + 2];

    if (half == 0) {                        // one writer per point
        #pragma unroll
        for (int i = 0; i < ORDER; ++i)
            sSx[p_local * ORDER + i] = sin2pi(x * (float)i);
    }

    float sy[ORDER];
    #pragma unroll
    for (int j = 0; j < ORDER; ++j)
        sy[j] = sin2pi(y * (float)j);

    // A-fragment lane layout needs only Sz[4m + 2*half + {0,1}] per lane.
    float szl[12];
    #pragma unroll
    for (int m = 0; m < 6; ++m) {
        szl[2 * m + 0] = sin2pi(z * (float)(4 * m + 2 * half + 0));
        szl[2 * m + 1] = sin2pi(z * (float)(4 * m + 2 * half + 1));
    }

    __syncthreads();

    // Zero-mask for i-tile-1 columns 24..31 (lm >= 8) -> keeps EXEC all-1s.
    const float himask = (lm < 8) ? 1.0f : 0.0f;
    // Per-lane element offset inside each jk window row of sCt.
    const int bbase = 2 * half * ORDER + lm;

    v8f acc0 = {};   // T[p, i=0..15]
    v8f acc1 = {};   // T[p, i=16..23] (cols 24..31 forced zero)

    // T[n,i] = sum_{jk} (Sy[n,j]*Sz[n,k]) * Ct[jk,i] ; K stepped 4 at a time.
    #pragma unroll
    for (int j = 0; j < ORDER; ++j) {
        const float syj = sy[j];
        #pragma unroll
        for (int m = 0; m < 6; ++m) {
            v2f a;
            a.x = syj * szl[2 * m + 0];    // A[lm][2*half]   in this K-window
            a.y = syj * szl[2 * m + 1];    // A[lm][2*half+1]

            const int base = (j * ORDER + 4 * m) * ORDER + bbase;
            v2f b0, b1;
            b0.x = sCt[base];                       // B[2*half  ][lm]
            b0.y = sCt[base + ORDER];               // B[2*half+1][lm]
            b1.x = sCt[base + 16] * himask;         // B[...][16+lm], padded tile
            b1.y = sCt[base + ORDER + 16] * himask;

            acc0 = __builtin_amdgcn_wmma_f32_16x16x4_f32(
                       false, a, false, b0, (short)0, acc0, false, false);
            acc1 = __builtin_amdgcn_wmma_f32_16x16x4_f32(
                       false, a, false, b1, (short)0, acc1, false, false);
        }
    }

    // ---- epilogue: out[p] = sum_i Sx[p][i] * T[p][i] ----
    // C/D layout: VGPR r holds row (point) r + 8*half, column i = lm.
    #pragma unroll
    for (int r = 0; r < 8; ++r) {
        const int pr = wave * PTS_PER_WAVE + r + 8 * half;   // local point
        float v = acc0[r] * sSx[pr * ORDER + lm];
        // acc1 lanes lm>=8 are exactly 0 (B was zeroed), so the padded read
        // (finite sin values / zeroed pad) contributes nothing.
        v += acc1[r] * sSx[pr * ORDER + 16 + lm];

        // Reduce the 16 i-columns within each 16-lane half (wave32 safe).
        v += __shfl_xor(v, 1, 32);
        v += __shfl_xor(v, 2, 32);
        v += __shfl_xor(v, 4, 32);
        v += __shfl_xor(v, 8, 32);

        if (lm == 0) {
            const int pg = blockIdx.x * PTS_PER_BLOCK + pr;
            if (pg < npts) out[pg] = v;
        }
    }
}

extern "C" void kernel_launch(void* const* d_in, const int* in_sizes, int n_in,
                              void* d_out, int out_size, void* d_ws, size_t ws_size,
                              hipStream_t stream) {
    const float* xyz    = (const float*)d_in[0];
    const float* coeffs = (const float*)d_in[1];
    float* out = (float*)d_out;
    const int npts   = in_sizes[0] / 3;
    const int blocks = (npts + PTS_PER_BLOCK - 1) / PTS_PER_BLOCK;
    FourierSurface_46119358824602_kernel<<<blocks, THREADS_PER_BLOCK, 0, stream>>>(
        xyz, coeffs, out, npts);
}